// CombinedLoss_33071248179545
// MI455X (gfx1250) — compile-verified
//
#include <hip/hip_runtime.h>

#ifndef __has_builtin
#define __has_builtin(x) 0
#endif

#define NUM_ANN 512
#define INF_F 100000000.0f

typedef float v4f __attribute__((ext_vector_type(4)));
typedef int   v4i __attribute__((ext_vector_type(4)));

// Exact pointer types the async-to-LDS builtin expects (from hipcc diagnostic:
// param 0 is "int __vector(4) __device__ *" == addrspace(1) v4i*).
typedef __attribute__((address_space(1))) v4i* gptr_v4i;
typedef __attribute__((address_space(3))) v4i* lptr_v4i;

#if __has_builtin(__builtin_amdgcn_global_load_async_to_lds_b128)
#define HAVE_ASYNC_LDS 1
#else
#define HAVE_ASYNC_LDS 0
#endif

__global__ __launch_bounds__(256) void fcos_assign_kernel(
    const float* __restrict__ ann,  // [512,3] (l, r, cls)
    const float* __restrict__ a1, const float* __restrict__ a2,
    const float* __restrict__ a3, const float* __restrict__ a4,
    const float* __restrict__ a5,
    float* __restrict__ out)        // [126976, 12]
{
    __shared__ float s_raw[NUM_ANN * 3];   // 6 KB raw staging
    __shared__ v4f   s_tab[NUM_ANN];       // 8 KB (l, r, limit, cls)

    const int tid = threadIdx.x;
    const int b   = blockIdx.x;

    // Level boundaries in points: 65536, 98304, 114688, 122880, 126976 — all
    // multiples of 256, so each block is entirely inside one level (uniform
    // stride/lo/hi in SGPRs). Constants folded at compile time in f64 then
    // rounded to f32, matching JAX weak-type promotion of python floats.
    constexpr double RATE = 22050.0 / 128.0;
    int level, pbase; const float* anch; float stride, lo, hi;
    if      (b < 256) { level=0; pbase=0;      anch=a1; stride=2.0f;  lo=(float)(-1.0*RATE);        hi=(float)(0.45608904*RATE);  }
    else if (b < 384) { level=1; pbase=65536;  anch=a2; stride=4.0f;  lo=(float)(0.45608904*RATE);  hi=(float)(0.878505635*RATE); }
    else if (b < 448) { level=2; pbase=98304;  anch=a3; stride=8.0f;  lo=(float)(0.878505635*RATE); hi=(float)(1.557724045*RATE); }
    else if (b < 480) { level=3; pbase=114688; anch=a4; stride=16.0f; lo=(float)(1.557724045*RATE); hi=(float)(2.264785525*RATE); }
    else              { level=4; pbase=122880; anch=a5; stride=32.0f; lo=(float)(2.264785525*RATE); hi=(float)(1000.0*RATE);      }

    const int gidx = b * 256 + tid;
    __builtin_prefetch(anch + (gidx - pbase), 0, 0);   // global_prefetch_b8

    // ---- Stage annotation table into LDS via CDNA5 async global->LDS copies.
    // 6144 bytes = 384 x b128; tracked by ASYNCcnt.
#if HAVE_ASYNC_LDS
    for (int k = tid; k < (NUM_ANN * 3) / 4; k += 256) {
        __builtin_amdgcn_global_load_async_to_lds_b128(
            (gptr_v4i)(ann + k * 4),
            (lptr_v4i)(s_raw + k * 4),
            /*offset=*/0, /*cpol=*/0);
    }
#if __has_builtin(__builtin_amdgcn_s_wait_asynccnt)
    __builtin_amdgcn_s_wait_asynccnt(0);
#else
    asm volatile("s_wait_asynccnt 0" ::: "memory");
#endif
#else
    for (int k = tid; k < NUM_ANN * 3; k += 256) s_raw[k] = ann[k];
#endif
    __syncthreads();

    // ---- Precompute per-annotation (l, r, limit, cls); limit folds the
    // per-level stride so the hot loop does no radius math.
    for (int j = tid; j < NUM_ANN; j += 256) {
        float l = s_raw[j * 3 + 0];
        float r = s_raw[j * 3 + 1];
        float c = s_raw[j * 3 + 2];
        float radius = ((c == 0.0f) ? 4.5f : 0.0f) + ((c == 1.0f) ? 1.5f : 0.0f);
        v4f t;
        t.x = l; t.y = r; t.z = l + radius * stride; t.w = c;
        s_tab[j] = t;
    }
    __syncthreads();

    const float p = anch[gidx - pbase];

    // ---- Masked argmin over 512 annotations. Strict '<' with best=INF,
    // bestIdx=0 reproduces jnp.argmin first-occurrence semantics (all-INF
    // row -> index 0). One broadcast ds_load_b128 + ~8 VALU per iteration.
    float best = INF_F;
    int bestIdx = 0;
#pragma unroll 4
    for (int j = 0; j < NUM_ANN; ++j) {
        v4f t = s_tab[j];
        float l = t.x, r = t.y, lim = t.z;
        float lstar = p - l;
        float rstar = r - p;
        float maxlr = fmaxf(lstar, rstar);
        bool valid = (p >= l) & (p <= fminf(r, lim)) & (maxlr >= lo) & (maxlr <= hi);
        float a = valid ? (r - l) : INF_F;
        if (a < best) { best = a; bestIdx = j; }
    }

    // ---- Epilogue: gather winner, emit 12 columns as three 16B stores.
    v4f t = s_tab[bestIdx];
    float l = t.x, r = t.y;
    float cls = (best == INF_F) ? 0.0f : t.w;
    float ls = p - l, rs = r - p;
    float inv = 1.0f / stride;           // stride is a power of two: exact
    float idxb = (bestIdx != 0) ? 1.0f : 0.0f;
    float lvl = (float)(level + 1);

    float* o = out + (size_t)gidx * 12;
    v4f o0; o0.x = idxb;      o0.y = l;        o0.z = r;        o0.w = cls;
    v4f o1; o1.x = l * inv;   o1.y = r * inv;  o1.z = cls;      o1.w = ls;
    v4f o2; o2.x = rs;        o2.y = ls * inv; o2.z = rs * inv; o2.w = lvl;
    *(v4f*)(o + 0) = o0;
    *(v4f*)(o + 4) = o1;
    *(v4f*)(o + 8) = o2;
}

extern "C" void kernel_launch(void* const* d_in, const int* in_sizes, int n_in,
                              void* d_out, int out_size, void* d_ws, size_t ws_size,
                              hipStream_t stream) {
    (void)in_sizes; (void)n_in; (void)out_size; (void)d_ws; (void)ws_size;
    const float* ann = (const float*)d_in[0];
    const float* a1  = (const float*)d_in[1];
    const float* a2  = (const float*)d_in[2];
    const float* a3  = (const float*)d_in[3];
    const float* a4  = (const float*)d_in[4];
    const float* a5  = (const float*)d_in[5];
    // 126976 points total / 256 threads = 496 blocks (level-aligned).
    fcos_assign_kernel<<<496, 256, 0, stream>>>(ann, a1, a2, a3, a4, a5,
                                                (float*)d_out);
}